// DeformationGNN_65841848647821
// MI455X (gfx1250) — compile-verified
//
#include <hip/hip_runtime.h>
#include <hip/hip_bf16.h>

typedef __attribute__((ext_vector_type(2))) float v2f;
typedef __attribute__((ext_vector_type(8))) float v8f;

#define D_HID 128
#define LRELU_SLOPE 0.2f

// ---------------- utility ----------------
__global__ void zero_f32_kernel(float* __restrict__ p, int n) {
    int i = blockIdx.x * blockDim.x + threadIdx.x;
    if (i < n) p[i] = 0.0f;
}

// ---------------- degree / norm ----------------
__global__ void degree_kernel(const long long* __restrict__ dst,
                              float* __restrict__ deg, int E) {
    int e = blockIdx.x * blockDim.x + threadIdx.x;
    if (e < E) atomicAdd(&deg[(int)dst[e]], 1.0f);
}

__global__ void dinv_kernel(float* __restrict__ deg, int n) {
    int i = blockIdx.x * blockDim.x + threadIdx.x;
    if (i < n) deg[i] = rsqrtf(deg[i] + 1.0f);  // deg>=1 always (self-loop)
}

// ---------------- layer-1 GEMM: H = X @ W  (M x 128 @ 128 x 128, fp32 WMMA) -------
// One wave computes one 16x16 output tile via 32 chained V_WMMA_F32_16X16X4_F32.
// A layout (16x4 f32): lanes 0-15 rows M=0..15 hold K=k0,k0+1 ; lanes 16-31 hold K=k0+2,k0+3
// B layout (4x16 f32): VGPR0 = {K=k0 n=0..15 | K=k0+2 n=0..15}, VGPR1 = K=k0+1 | k0+3
// C/D (16x16 f32): VGPR r: lanes 0-15 -> M=r, lanes 16-31 -> M=r+8
__global__ __launch_bounds__(128)
void gemm128_wmma_f32(const float* __restrict__ A, const float* __restrict__ B,
                      float* __restrict__ C, int M) {
    const int N = 128, K = 128;
    int tm   = blockIdx.x;                       // M-tile (16 rows)
    int tn   = blockIdx.y * 4 + threadIdx.y;     // N-tile (16 cols), 4 waves/block
    int lane = threadIdx.x;                      // 0..31
    int half = lane >> 4;                        // 0: K lo pair, 1: K hi pair
    int l15  = lane & 15;

    const float* arow = A + (size_t)(tm * 16 + l15) * K + 2 * half;
    const float* bcol = B + (size_t)(2 * half) * N + (size_t)tn * 16 + l15;

    v8f acc = {};
#pragma unroll 8
    for (int k0 = 0; k0 < K; k0 += 4) {
        v2f a, b;
        a.x = arow[k0];
        a.y = arow[k0 + 1];
        b.x = bcol[(size_t)k0 * N];
        b.y = bcol[(size_t)(k0 + 1) * N];
        acc = __builtin_amdgcn_wmma_f32_16x16x4_f32(
            /*neg_a=*/false, a, /*neg_b=*/false, b,
            /*c_mod=*/(short)0, acc, /*reuse_a=*/false, /*reuse_b=*/false);
    }

    float* crow = C + (size_t)(tm * 16 + half * 8) * N + (size_t)tn * 16 + l15;
#pragma unroll
    for (int r = 0; r < 8; ++r) crow[(size_t)r * N] = acc[r];
}

// ---------------- layer-1 edge aggregation: agg[dst] += h1[src]*norm -------------
// 32 lanes per edge; each lane handles 4 consecutive features (float4 gather).
__global__ void agg128_kernel(const long long* __restrict__ src,
                              const long long* __restrict__ dst,
                              const float* __restrict__ dinv,
                              const float* __restrict__ h1,
                              float* __restrict__ agg, int E) {
    long long t = (long long)blockIdx.x * blockDim.x + threadIdx.x;
    int e    = (int)(t >> 5);
    int lane = (int)(t & 31);
    if (e >= E) return;
    int s = (int)src[e], d = (int)dst[e];
    float norm = dinv[s] * dinv[d];
    float4 hv  = ((const float4*)(h1 + (size_t)s * D_HID))[lane];
    float* out = agg + (size_t)d * D_HID + lane * 4;
    atomicAdd(out + 0, hv.x * norm);
    atomicAdd(out + 1, hv.y * norm);
    atomicAdd(out + 2, hv.z * norm);
    atomicAdd(out + 3, hv.w * norm);
}

// ---------------- self-loop + bias + leaky relu (in place on agg) ----------------
__global__ void act_kernel(float* __restrict__ agg, const float* __restrict__ h1,
                           const float* __restrict__ dinv, const float* __restrict__ b1,
                           int total) {
    int t = blockIdx.x * blockDim.x + threadIdx.x;
    if (t >= total) return;
    int i = t >> 7, f = t & 127;
    float di = dinv[i];
    float v  = agg[t] + h1[t] * di * di + b1[f];
    agg[t]   = v > 0.0f ? v : LRELU_SLOPE * v;
}

// ---------------- layer-2 GEMM: H2 = Hact @ W2  (128 -> 3), one wave per node ----
__global__ __launch_bounds__(256)
void gemm3_kernel(const float* __restrict__ Hact, const float* __restrict__ W2,
                  float* __restrict__ H2, int M) {
    int gt   = blockIdx.x * blockDim.x + threadIdx.x;
    int node = gt >> 5;
    int lane = gt & 31;
    if (node >= M) return;
    float4 h = ((const float4*)(Hact + (size_t)node * D_HID))[lane];
    int k = lane * 4;
    // W2 row-major 128x3
    float a0 = h.x * W2[k * 3 + 0] + h.y * W2[(k + 1) * 3 + 0] +
               h.z * W2[(k + 2) * 3 + 0] + h.w * W2[(k + 3) * 3 + 0];
    float a1 = h.x * W2[k * 3 + 1] + h.y * W2[(k + 1) * 3 + 1] +
               h.z * W2[(k + 2) * 3 + 1] + h.w * W2[(k + 3) * 3 + 1];
    float a2 = h.x * W2[k * 3 + 2] + h.y * W2[(k + 1) * 3 + 2] +
               h.z * W2[(k + 2) * 3 + 2] + h.w * W2[(k + 3) * 3 + 2];
#pragma unroll
    for (int off = 16; off > 0; off >>= 1) {
        a0 += __shfl_down(a0, off, 32);
        a1 += __shfl_down(a1, off, 32);
        a2 += __shfl_down(a2, off, 32);
    }
    if (lane == 0) {
        H2[(size_t)node * 3 + 0] = a0;
        H2[(size_t)node * 3 + 1] = a1;
        H2[(size_t)node * 3 + 2] = a2;
    }
}

// ---------------- layer-2 edge aggregation (3 feats/edge) ------------------------
__global__ void agg3_kernel(const long long* __restrict__ src,
                            const long long* __restrict__ dst,
                            const float* __restrict__ dinv,
                            const float* __restrict__ h2,
                            float* __restrict__ agg, int E) {
    int e = blockIdx.x * blockDim.x + threadIdx.x;
    if (e >= E) return;
    int s = (int)src[e], d = (int)dst[e];
    float norm = dinv[s] * dinv[d];
    atomicAdd(&agg[(size_t)d * 3 + 0], h2[(size_t)s * 3 + 0] * norm);
    atomicAdd(&agg[(size_t)d * 3 + 1], h2[(size_t)s * 3 + 1] * norm);
    atomicAdd(&agg[(size_t)d * 3 + 2], h2[(size_t)s * 3 + 2] * norm);
}

// ---------------- final epilogue -------------------------------------------------
__global__ void final_kernel(const float* __restrict__ agg2, const float* __restrict__ h2,
                             const float* __restrict__ dinv, const float* __restrict__ b2,
                             float* __restrict__ out, int total) {
    int t = blockIdx.x * blockDim.x + threadIdx.x;
    if (t >= total) return;
    int i = t / 3, j = t - i * 3;
    float di = dinv[i];
    out[t] = agg2[t] + h2[t] * di * di + b2[j];
}

extern "C" void kernel_launch(void* const* d_in, const int* in_sizes, int n_in,
                              void* d_out, int out_size, void* d_ws, size_t ws_size,
                              hipStream_t stream) {
    const float*     X    = (const float*)d_in[0];      // [N,128]
    const float*     W1   = (const float*)d_in[1];      // [128,128]
    const float*     b1   = (const float*)d_in[2];      // [128]
    const float*     W2   = (const float*)d_in[3];      // [128,3]
    const float*     b2   = (const float*)d_in[4];      // [3]
    const long long* ei   = (const long long*)d_in[5];  // [2,E] int64
    float*           out  = (float*)d_out;

    const int M = in_sizes[0] / D_HID;      // 100000 nodes (multiple of 16)
    const int E = in_sizes[5] / 2;          // 1,600,000 edges
    const long long* src = ei;
    const long long* dst = ei + E;

    // workspace layout (256B aligned)
    char* ws = (char*)d_ws;
    size_t off = 0;
    auto alloc = [&](size_t bytes) {
        char* p = ws + off;
        off += (bytes + 255) & ~(size_t)255;
        return p;
    };
    float* dinv = (float*)alloc((size_t)M * sizeof(float));            // deg -> dinv
    float* h1   = (float*)alloc((size_t)M * D_HID * sizeof(float));    // X@W1
    float* agg1 = (float*)alloc((size_t)M * D_HID * sizeof(float));    // aggregated, then activated
    float* h2   = (float*)alloc((size_t)M * 3 * sizeof(float));        // Hact@W2
    float* agg2 = (float*)alloc((size_t)M * 3 * sizeof(float));        // aggregated
    (void)ws_size;

    const int TB = 256;
    auto blocks = [&](long long n) { return (unsigned)((n + TB - 1) / TB); };

    // 1) zero accumulators (ws is poisoned, must re-zero every call)
    zero_f32_kernel<<<blocks(M), TB, 0, stream>>>(dinv, M);
    zero_f32_kernel<<<blocks((long long)M * D_HID), TB, 0, stream>>>(agg1, M * D_HID);
    zero_f32_kernel<<<blocks((long long)M * 3), TB, 0, stream>>>(agg2, M * 3);

    // 2) degrees and symmetric-norm factors
    degree_kernel<<<blocks(E), TB, 0, stream>>>(dst, dinv, E);
    dinv_kernel<<<blocks(M), TB, 0, stream>>>(dinv, M);

    // 3) layer-1 GEMM via fp32 WMMA: h1 = X @ W1
    {
        dim3 grid(M / 16, D_HID / 16 / 4);
        dim3 block(32, 4);
        gemm128_wmma_f32<<<grid, block, 0, stream>>>(X, W1, h1, M);
    }

    // 4) layer-1 edge aggregation (bandwidth-dominant stage)
    agg128_kernel<<<blocks((long long)E * 32), TB, 0, stream>>>(src, dst, dinv, h1, agg1, E);

    // 5) self-loop + bias + leaky relu (in place -> activated features in agg1)
    act_kernel<<<blocks((long long)M * D_HID), TB, 0, stream>>>(agg1, h1, dinv, b1, M * D_HID);

    // 6) layer-2 GEMM (128 -> 3), one wave32 per node
    gemm3_kernel<<<blocks((long long)M * 32), TB, 0, stream>>>(agg1, W2, h2, M);

    // 7) layer-2 edge aggregation
    agg3_kernel<<<blocks(E), TB, 0, stream>>>(src, dst, dinv, h2, agg2, E);

    // 8) final epilogue -> d_out
    final_kernel<<<blocks((long long)M * 3), TB, 0, stream>>>(agg2, h2, dinv, b2, out, M * 3);
}